// LabelSmoothingLoss_50611894616521
// MI455X (gfx1250) — compile-verified
//
#include <hip/hip_runtime.h>
#include <cstdint>
#include <cstddef>

// LabelSmoothingLoss for MI455X (gfx1250).
// One workgroup per row; row staged HBM -> LDS exactly once via CDNA5
// async global->LDS DMA (global_load_async_to_lds_b128, ASYNCcnt).
// Then max / sum / sum-exp passes and the 100-label gather run out of LDS.
// HBM traffic: 262 MB read once => ~11.3 us at 23.3 TB/s (roofline-bound).

#ifndef VOCAB
#define VOCAB 32000
#endif

namespace {

constexpr int   kV      = VOCAB;        // vocabulary size
constexpr int   kL      = 50;           // labels per half
constexpr int   kTPB    = 1024;         // 32 waves (wave32)
constexpr int   kChunks = kV / 4;       // 8000 float4 chunks per row
constexpr float kEps    = 0.1f;

__device__ __forceinline__ float wred_max(float v) {
#pragma unroll
  for (int o = 16; o > 0; o >>= 1) v = fmaxf(v, __shfl_xor(v, o, 32));
  return v;
}
__device__ __forceinline__ float wred_sum(float v) {
#pragma unroll
  for (int o = 16; o > 0; o >>= 1) v += __shfl_xor(v, o, 32);
  return v;
}

__global__ void __launch_bounds__(kTPB, 1)
ls_loss_kernel(const float* __restrict__ output,
               const int*   __restrict__ label,
               const float* __restrict__ score,
               const int*   __restrict__ tlabel,
               const float* __restrict__ tscore,
               float*       __restrict__ loss)
{
  extern __shared__ float sm[];          // [kV row][128 scratch]
  float* row = sm;
  float* scr = sm + kV;

  const int b   = blockIdx.x;
  const int tid = threadIdx.x;
  const int lane = tid & 31;
  const int wid  = tid >> 5;

  // ---------------------------------------------------------------
  // Stage row b (125 KB) into LDS with async global->LDS DMA.
  // GVS mode: SGPR64 base + per-lane 32-bit byte offset.
  // Trip count is uniform (full EXEC); tail lanes clamp to the last
  // chunk and redundantly rewrite identical bytes.
  // ---------------------------------------------------------------
  const unsigned long long gbase =
      (unsigned long long)(const void*)(output + (size_t)b * kV);
  const unsigned int ldsbase = (unsigned int)(uintptr_t)row;

  for (int i = tid; i < 8192; i += kTPB) {
    const int j = (i < kChunks) ? i : (kChunks - 1);
    const unsigned int off = (unsigned int)j * 16u;
    asm volatile("global_load_async_to_lds_b128 %0, %1, %2"
                 :
                 : "v"(ldsbase + off), "v"(off), "s"(gbase)
                 : "memory");
  }
  asm volatile("s_wait_asynccnt 0" ::: "memory");
  __syncthreads();

  // ---------------------------------------------------------------
  // Pass A: row max and raw sum (for sum of log-probs), from LDS.
  // ---------------------------------------------------------------
  const float4* row4 = reinterpret_cast<const float4*>(row);
  float lmax = -3.402823466e38f;
  float lsum = 0.0f;
  for (int i = tid; i < kChunks; i += kTPB) {
    const float4 v = row4[i];
    lmax = fmaxf(lmax, fmaxf(fmaxf(v.x, v.y), fmaxf(v.z, v.w)));
    lsum += (v.x + v.y) + (v.z + v.w);
  }
  lmax = wred_max(lmax);
  lsum = wred_sum(lsum);
  if (lane == 0) { scr[wid] = lmax; scr[32 + wid] = lsum; }
  __syncthreads();
  if (tid < 32) {
    float m = wred_max(scr[tid]);
    float s = wred_sum(scr[32 + tid]);
    if (tid == 0) { scr[64] = m; scr[65] = s; }
  }
  __syncthreads();
  const float rmax = scr[64];
  const float rsum = scr[65];

  // ---------------------------------------------------------------
  // Pass B: sum of exp(x - max), from LDS.
  // ---------------------------------------------------------------
  float lexp = 0.0f;
  for (int i = tid; i < kChunks; i += kTPB) {
    const float4 v = row4[i];
    lexp += __expf(v.x - rmax) + __expf(v.y - rmax) +
            __expf(v.z - rmax) + __expf(v.w - rmax);
  }
  lexp = wred_sum(lexp);
  if (lane == 0) scr[wid] = lexp;   // scr[0..31] free again (all passed barrier)
  __syncthreads();
  if (tid < 32) {
    float s = wred_sum(scr[tid]);
    if (tid == 0) scr[66] = s;
  }
  __syncthreads();
  const float lse = rmax + __logf(scr[66]);

  // ---------------------------------------------------------------
  // Gather the <=100 label positions (t==0 is padding -> dropped).
  // Deterministic reduction (no atomics) for bit-stable replays.
  // ---------------------------------------------------------------
  float contrib = 0.0f;
  if (tid < 2 * kL) {
    int   t;
    float s;
    if (tid < kL) {                                 // test_label / test_score
      t = tlabel[(size_t)b * kL + tid];
      s = tscore[(size_t)b * kL + tid];
    } else {                                        // label / score
      t = label[(size_t)b * kL + (tid - kL)];
      s = score[(size_t)b * kL + (tid - kL)];
    }
    if (t != 0) contrib = s * (row[t - 1] - lse);
  }
  contrib = wred_sum(contrib);
  if (lane == 0) scr[wid] = contrib;
  __syncthreads();

  if (tid == 0) {
    float acc = 0.0f;
#pragma unroll
    for (int w = 0; w < 4; ++w) acc += scr[w];      // only warps 0..3 nonzero
    const float epsv   = kEps / (float)kV;
    const float coef   = 1.0f - kEps - epsv;
    const float sum_lp = rsum - (float)kV * lse;
    loss[b] = -(epsv * sum_lp + coef * acc);
  }
}

} // namespace

extern "C" void kernel_launch(void* const* d_in, const int* in_sizes, int n_in,
                              void* d_out, int out_size, void* d_ws, size_t ws_size,
                              hipStream_t stream) {
  (void)n_in; (void)d_ws; (void)ws_size; (void)out_size;

  const float* output = (const float*)d_in[0];   // (B, V) f32
  const int*   label  = (const int*)  d_in[1];   // (B, L) i32
  const float* score  = (const float*)d_in[2];   // (B, L) f32
  const int*   tlabel = (const int*)  d_in[3];   // (B, L) i32
  const float* tscore = (const float*)d_in[4];   // (B, L) f32
  float*       loss   = (float*)d_out;           // (B,)   f32

  const int rows = in_sizes[0] / kV;             // B = 2048
  const size_t shmem = (size_t)(kV + 128) * sizeof(float);  // 128.5 KB

  // Allow >64KB dynamic LDS (320KB/WGP on CDNA5). Idempotent; safe
  // under graph capture (not a stream-ordered op). Ignore errors.
  (void)hipFuncSetAttribute((const void*)ls_loss_kernel,
                            hipFuncAttributeMaxDynamicSharedMemorySize,
                            (int)shmem);

  ls_loss_kernel<<<rows, kTPB, shmem, stream>>>(output, label, score,
                                                tlabel, tscore, loss);
}